// EdgeClassifierGAT_1949915152970
// MI455X (gfx1250) — compile-verified
//
#include <hip/hip_runtime.h>
#include <math.h>

#define N_NODES 50000
#define N_EDGES 600000
#define IN_F    128
#define HID     64
#define HEADS   4
#define ED      16
#define OUT_F   2
#define C1      256   // HEADS*HID
#define NEG_SLOPE 0.2f

typedef float v2f __attribute__((ext_vector_type(2)));
typedef float v8f __attribute__((ext_vector_type(8)));
typedef unsigned int u32x4 __attribute__((ext_vector_type(4)));
typedef int i32x4 __attribute__((ext_vector_type(4)));
typedef int i32x8 __attribute__((ext_vector_type(8)));

#if defined(__HIP_DEVICE_COMPILE__) && __has_builtin(__builtin_amdgcn_tensor_load_to_lds)
#define HAVE_TDM 1
#else
#define HAVE_TDM 0
#endif

#if HAVE_TDM
// ---------------------------------------------------------------------------
// Tensor Data Mover: 2D tile load Global -> LDS per CDNA5 ISA D# layout (§8).
// tensor dims == tile dims (tiles always fully in-bounds here).
// Optional LDS row padding done by the DMA engine itself (pad fields).
// ---------------------------------------------------------------------------
__device__ __forceinline__ void tdm_load_2d(const void* gptr, unsigned lds_off,
                                            unsigned tile_w, unsigned tile_h,
                                            unsigned long long stride_elems,
                                            unsigned pad_en, unsigned pad_interval,
                                            unsigned pad_amount) {
    unsigned long long ga = (unsigned long long)(size_t)gptr;
    u32x4 g0;
    g0[0] = 1u;                                            // count=1, user descriptor
    g0[1] = lds_off;                                       // lds_addr (bytes)
    g0[2] = (unsigned)ga;                                  // global_addr[31:0]
    g0[3] = (unsigned)((ga >> 32) & 0x01FFFFFFu) | (2u << 30);  // ga[56:32] | type=2
    i32x8 g1;
    g1[0] = (int)((2u << 16) | (pad_en << 20) | (pad_interval << 22) | (pad_amount << 25));
    g1[1] = (int)((tile_w & 0xFFFFu) << 16);               // tensor_dim0[15:0]
    g1[2] = (int)((tile_w >> 16) | ((tile_h & 0xFFFFu) << 16));  // td0 hi | td1 lo
    g1[3] = (int)((tile_h >> 16) | ((tile_w & 0xFFFFu) << 16));  // td1 hi | tile_dim0
    g1[4] = (int)(tile_h & 0xFFFFu);                       // tile_dim1 | tile_dim2=0
    g1[5] = (int)(unsigned)(stride_elems & 0xFFFFFFFFull); // tensor_dim0_stride lo
    g1[6] = (int)(unsigned)(stride_elems >> 32);           // stride hi | d1stride lo = 0
    g1[7] = 0;
    i32x4 z4 = {0, 0, 0, 0};
#if __clang_major__ >= 23
    i32x8 z8 = {0, 0, 0, 0, 0, 0, 0, 0};
    __builtin_amdgcn_tensor_load_to_lds(g0, g1, z4, z4, z8, 0);
#else
    __builtin_amdgcn_tensor_load_to_lds(g0, g1, z4, z4, 0);
#endif
}
#define TDM_WAIT() __builtin_amdgcn_s_wait_tensorcnt(0)
#define LDS_OFF(p) ((unsigned)(size_t)(const void*)(p))
#endif

// ---------- helpers ----------
__device__ __forceinline__ void atomicMaxF(float* addr, float val) {
    // monotone int-bit encoding of IEEE754; valid for mixed signs given -inf init
    if (val >= 0.0f) {
        atomicMax((int*)addr, __float_as_int(val));
    } else {
        atomicMin((unsigned int*)addr, (unsigned int)__float_as_int(val));
    }
}

__global__ void fill_f32(float* __restrict__ p, float v, size_t n) {
    size_t i = (size_t)blockIdx.x * blockDim.x + threadIdx.x;
    if (i < n) p[i] = v;
}

// ---------------------------------------------------------------------------
// WMMA f32 GEMM: C[M,Nc] = A[M,K] @ W[K,Nc] + bias
// Block = 4 waves = 16(M) x 64(N) tile. K staged in 32-wide chunks through LDS
// by the TDM (double-buffered: DMA of chunk c+1 overlaps WMMA on chunk c).
// TDM pads A rows to 33 DWORDs and B rows to 65 DWORDs for bank spread.
// ---------------------------------------------------------------------------
__global__ void gemm_bias_wmma(const float* __restrict__ A, const float* __restrict__ W,
                               const float* __restrict__ bias, float* __restrict__ C,
                               int M, int K, int Nc) {
    constexpr int KC = 32;
    __shared__ float sA[2][16][KC + 1];
    __shared__ float sB[2][KC][64 + 1];

    int wave = threadIdx.x >> 5;
    int lane = threadIdx.x & 31;
    int kh   = lane >> 4;
    int l15  = lane & 15;
    int m0    = blockIdx.x * 16;
    int n0blk = blockIdx.y * 64;
    int ncol  = wave * 16 + l15;    // column within the 64-wide block tile
    int nchunk = K / KC;

    v8f acc = {};

#if HAVE_TDM
    if (threadIdx.x < 32) {
        tdm_load_2d(A + (size_t)m0 * K, LDS_OFF(&sA[0][0][0]), KC, 16,
                    (unsigned long long)K, 1u, 4u, 0u);          // +1 dword / 32
        tdm_load_2d(W + n0blk, LDS_OFF(&sB[0][0][0]), 64, KC,
                    (unsigned long long)Nc, 1u, 5u, 0u);         // +1 dword / 64
        TDM_WAIT();
    }
    __syncthreads();

    for (int c = 0; c < nchunk; ++c) {
        int buf = c & 1;
        if (threadIdx.x < 32 && c + 1 < nchunk) {
            int k0 = (c + 1) * KC;
            tdm_load_2d(A + (size_t)m0 * K + k0, LDS_OFF(&sA[buf ^ 1][0][0]), KC, 16,
                        (unsigned long long)K, 1u, 4u, 0u);
            tdm_load_2d(W + (size_t)k0 * Nc + n0blk, LDS_OFF(&sB[buf ^ 1][0][0]), 64, KC,
                        (unsigned long long)Nc, 1u, 5u, 0u);
        }
        #pragma unroll
        for (int k = 0; k < KC; k += 4) {
            v2f a, b;
            a.x = sA[buf][l15][k + kh * 2];
            a.y = sA[buf][l15][k + kh * 2 + 1];
            b.x = sB[buf][k + kh * 2][ncol];
            b.y = sB[buf][k + kh * 2 + 1][ncol];
            acc = __builtin_amdgcn_wmma_f32_16x16x4_f32(false, a, false, b,
                                                        (short)0, acc, false, false);
        }
        if (threadIdx.x < 32 && c + 1 < nchunk) TDM_WAIT();
        __syncthreads();
    }
#else
    for (int c = 0; c < nchunk; ++c) {
        int k0 = c * KC;
        for (int idx = threadIdx.x; idx < 16 * KC; idx += blockDim.x) {
            int r = idx >> 5, kk = idx & 31;
            sA[0][r][kk] = A[(size_t)(m0 + r) * K + k0 + kk];
        }
        for (int idx = threadIdx.x; idx < KC * 64; idx += blockDim.x) {
            int r = idx >> 6, nn = idx & 63;
            sB[0][r][nn] = W[(size_t)(k0 + r) * Nc + n0blk + nn];
        }
        __syncthreads();
        #pragma unroll
        for (int k = 0; k < KC; k += 4) {
            v2f a, b;
            a.x = sA[0][l15][k + kh * 2];
            a.y = sA[0][l15][k + kh * 2 + 1];
            b.x = sB[0][k + kh * 2][ncol];
            b.y = sB[0][k + kh * 2 + 1][ncol];
            acc = __builtin_amdgcn_wmma_f32_16x16x4_f32(false, a, false, b,
                                                        (short)0, acc, false, false);
        }
        __syncthreads();
    }
#endif

    float bv = bias[n0blk + wave * 16 + l15];
    int rbase = m0 + kh * 8;
    #pragma unroll
    for (int v = 0; v < 8; ++v)
        C[(size_t)(rbase + v) * Nc + n0blk + wave * 16 + l15] = acc[v] + bv;
}

// ---------- layer-1 edge logits: one wave per edge, 256 channels, 4 heads ----------
__global__ void edge_logits1(const float* __restrict__ xl, const float* __restrict__ xr,
                             const float* __restrict__ edge_attr,
                             const int* __restrict__ src, const int* __restrict__ dst,
                             const float* __restrict__ We, const float* __restrict__ be,
                             const float* __restrict__ att,
                             float* __restrict__ logits, float* __restrict__ amax) {
    __shared__ float sW[ED * C1];
    __shared__ float sB[C1];
    __shared__ float sAtt[C1];
#if HAVE_TDM
    if (threadIdx.x < 32) tdm_load_2d(We, LDS_OFF(sW), ED * C1, 1,
                                      (unsigned long long)(ED * C1), 0u, 0u, 0u);
#else
    for (int i = threadIdx.x; i < ED * C1; i += blockDim.x) sW[i] = We[i];
#endif
    for (int i = threadIdx.x; i < C1; i += blockDim.x) { sB[i] = be[i]; sAtt[i] = att[i]; }
#if HAVE_TDM
    if (threadIdx.x < 32) TDM_WAIT();
#endif
    __syncthreads();

    int wave = threadIdx.x >> 5;
    int lane = threadIdx.x & 31;
    int e = blockIdx.x * (blockDim.x >> 5) + wave;
    if (e >= N_EDGES) return;

    int s = src[e], d = dst[e];
    float ea[ED];
    #pragma unroll
    for (int j = 0; j < ED; ++j) ea[j] = edge_attr[(size_t)e * ED + j];

    float p[HEADS] = {0.f, 0.f, 0.f, 0.f};
    #pragma unroll
    for (int j = 0; j < 8; ++j) {
        int c = lane + 32 * j;            // head = j>>1 for lane<32
        float xe = sB[c];
        #pragma unroll
        for (int q = 0; q < ED; ++q) xe += ea[q] * sW[q * C1 + c];
        float m = xl[(size_t)s * C1 + c] + xr[(size_t)d * C1 + c] + xe;
        m = (m > 0.f) ? m : NEG_SLOPE * m;
        p[j >> 1] += m * sAtt[c];
    }
    #pragma unroll
    for (int h = 0; h < HEADS; ++h)
        #pragma unroll
        for (int off = 16; off > 0; off >>= 1)
            p[h] += __shfl_xor(p[h], off, 32);

    if (lane < HEADS) {
        float v = p[lane];
        logits[(size_t)e * HEADS + lane] = v;
        atomicMaxF(&amax[(size_t)d * HEADS + lane], v);
    }
}

// ---------- layer-2 edge logits: one wave per edge, 64 channels, 1 head ----------
__global__ void edge_logits2(const float* __restrict__ xl, const float* __restrict__ xr,
                             const float* __restrict__ edge_attr,
                             const int* __restrict__ src, const int* __restrict__ dst,
                             const float* __restrict__ We, const float* __restrict__ be,
                             const float* __restrict__ att,
                             float* __restrict__ logits, float* __restrict__ amax) {
    __shared__ float sW[ED * HID];
    __shared__ float sB[HID];
    __shared__ float sAtt[HID];
#if HAVE_TDM
    if (threadIdx.x < 32) tdm_load_2d(We, LDS_OFF(sW), ED * HID, 1,
                                      (unsigned long long)(ED * HID), 0u, 0u, 0u);
#else
    for (int i = threadIdx.x; i < ED * HID; i += blockDim.x) sW[i] = We[i];
#endif
    for (int i = threadIdx.x; i < HID; i += blockDim.x) { sB[i] = be[i]; sAtt[i] = att[i]; }
#if HAVE_TDM
    if (threadIdx.x < 32) TDM_WAIT();
#endif
    __syncthreads();

    int wave = threadIdx.x >> 5;
    int lane = threadIdx.x & 31;
    int e = blockIdx.x * (blockDim.x >> 5) + wave;
    if (e >= N_EDGES) return;

    int s = src[e], d = dst[e];
    float ea[ED];
    #pragma unroll
    for (int j = 0; j < ED; ++j) ea[j] = edge_attr[(size_t)e * ED + j];

    float p = 0.f;
    #pragma unroll
    for (int j = 0; j < 2; ++j) {
        int c = lane + 32 * j;
        float xe = sB[c];
        #pragma unroll
        for (int q = 0; q < ED; ++q) xe += ea[q] * sW[q * HID + c];
        float m = xl[(size_t)s * HID + c] + xr[(size_t)d * HID + c] + xe;
        m = (m > 0.f) ? m : NEG_SLOPE * m;
        p += m * sAtt[c];
    }
    #pragma unroll
    for (int off = 16; off > 0; off >>= 1) p += __shfl_xor(p, off, 32);

    if (lane == 0) {
        logits[e] = p;
        atomicMaxF(&amax[d], p);
    }
}

// ---------- softmax numerator + denominator accumulation ----------
__global__ void softmax_ex(float* __restrict__ logits, const int* __restrict__ dst,
                           const float* __restrict__ amax, float* __restrict__ denom,
                           int E, int H) {
    size_t idx = (size_t)blockIdx.x * blockDim.x + threadIdx.x;
    if (idx >= (size_t)E * H) return;
    int e = (int)(idx / H);
    int h = (int)(idx - (size_t)e * H);
    int d = dst[e];
    float ex = __expf(logits[idx] - amax[(size_t)d * H + h]);
    logits[idx] = ex;                       // overwrite logits with exp numerator
    atomicAdd(&denom[(size_t)d * H + h], ex);
}

// ---------- weighted scatter-add aggregation ----------
__global__ void aggregate(const float* __restrict__ xl, const float* __restrict__ ex,
                          const float* __restrict__ denom,
                          const int* __restrict__ src, const int* __restrict__ dst,
                          float* __restrict__ hout, int E, int C, int ch) {
    size_t idx = (size_t)blockIdx.x * blockDim.x + threadIdx.x;
    if (idx >= (size_t)E * C) return;
    int e = (int)(idx / C);
    int c = (int)(idx - (size_t)e * C);
    int h = c / ch;
    int H = C / ch;
    int s = src[e], d = dst[e];
    float alpha = ex[(size_t)e * H + h] / denom[(size_t)d * H + h];
    atomicAdd(&hout[(size_t)d * C + c], xl[(size_t)s * C + c] * alpha);
}

__global__ void add_bias_relu(float* __restrict__ h, const float* __restrict__ bias,
                              int C, size_t n) {
    size_t i = (size_t)blockIdx.x * blockDim.x + threadIdx.x;
    if (i >= n) return;
    int c = (int)(i % C);
    float v = h[i] + bias[c];
    h[i] = v > 0.f ? v : 0.f;
}

// ---------------------------------------------------------------------------
// edge MLP: 16 edges per block; Wm1 (144x64 = 36KB) staged by TDM while the
// block gathers [h2[src], h2[dst], edge_attr] into LDS; then WMMA over K=144.
// ---------------------------------------------------------------------------
__global__ void edge_mlp(const float* __restrict__ h2, const float* __restrict__ edge_attr,
                         const int* __restrict__ src, const int* __restrict__ dst,
                         const float* __restrict__ Wm1, const float* __restrict__ bm1,
                         const float* __restrict__ Wm2, const float* __restrict__ bm2,
                         float* __restrict__ out) {
    const int KF = 2 * HID + ED;            // 144
    __shared__ float sW1[144 * HID];        // 36 KB
    __shared__ float sA[16][144 + 4];       // gathered edge features (padded)
    __shared__ float sH[16][HID];           // relu(hidden)

#if HAVE_TDM
    if (threadIdx.x < 32)
        tdm_load_2d(Wm1, LDS_OFF(sW1), KF * HID, 1,
                    (unsigned long long)(KF * HID), 0u, 0u, 0u);
#else
    for (int i = threadIdx.x; i < KF * HID; i += blockDim.x) sW1[i] = Wm1[i];
#endif

    int e0 = blockIdx.x * 16;
    for (int idx = threadIdx.x; idx < 16 * KF; idx += blockDim.x) {
        int r = idx / KF, c = idx - r * KF;
        int e = e0 + r;
        float v;
        if (c < HID)            v = h2[(size_t)src[e] * HID + c];
        else if (c < 2 * HID)   v = h2[(size_t)dst[e] * HID + (c - HID)];
        else                    v = edge_attr[(size_t)e * ED + (c - 2 * HID)];
        sA[r][c] = v;
    }
#if HAVE_TDM
    if (threadIdx.x < 32) TDM_WAIT();
#endif
    __syncthreads();

    int wave = threadIdx.x >> 5;
    int lane = threadIdx.x & 31;
    int kh   = lane >> 4;
    int l15  = lane & 15;
    int n0 = wave * 16;                     // 4 waves cover 64 output channels

    v8f acc = {};
    for (int k = 0; k < KF; k += 4) {
        v2f a, b;
        a.x = sA[l15][k + kh * 2];
        a.y = sA[l15][k + kh * 2 + 1];
        b.x = sW1[(size_t)(k + kh * 2) * HID + n0 + l15];
        b.y = sW1[(size_t)(k + kh * 2 + 1) * HID + n0 + l15];
        acc = __builtin_amdgcn_wmma_f32_16x16x4_f32(false, a, false, b,
                                                    (short)0, acc, false, false);
    }
    float bv = bm1[n0 + l15];
    #pragma unroll
    for (int v = 0; v < 8; ++v) {
        float hv = acc[v] + bv;
        sH[kh * 8 + v][n0 + l15] = hv > 0.f ? hv : 0.f;
    }
    __syncthreads();

    if (threadIdx.x < 32) {
        int r = threadIdx.x >> 1;
        int o = threadIdx.x & 1;
        float sum = bm2[o];
        #pragma unroll
        for (int c = 0; c < HID; ++c) sum += sH[r][c] * Wm2[c * OUT_F + o];
        out[(size_t)(e0 + r) * OUT_F + o] = sum;
    }
}

// ---------- host launch ----------
extern "C" void kernel_launch(void* const* d_in, const int* in_sizes, int n_in,
                              void* d_out, int out_size, void* d_ws, size_t ws_size,
                              hipStream_t stream) {
    const float* x        = (const float*)d_in[0];
    const int*   ei       = (const int*)d_in[1];
    const float* eattr    = (const float*)d_in[2];
    const float* W1l = (const float*)d_in[3];  const float* b1l = (const float*)d_in[4];
    const float* W1r = (const float*)d_in[5];  const float* b1r = (const float*)d_in[6];
    const float* W1e = (const float*)d_in[7];  const float* b1e = (const float*)d_in[8];
    const float* att1 = (const float*)d_in[9]; const float* bias1 = (const float*)d_in[10];
    const float* W2l = (const float*)d_in[11]; const float* b2l = (const float*)d_in[12];
    const float* W2r = (const float*)d_in[13]; const float* b2r = (const float*)d_in[14];
    const float* W2e = (const float*)d_in[15]; const float* b2e = (const float*)d_in[16];
    const float* att2 = (const float*)d_in[17]; const float* bias2 = (const float*)d_in[18];
    const float* Wm1 = (const float*)d_in[19]; const float* bm1 = (const float*)d_in[20];
    const float* Wm2 = (const float*)d_in[21]; const float* bm2 = (const float*)d_in[22];
    float* out = (float*)d_out;

    const int* src = ei;
    const int* dst = ei + N_EDGES;

    // workspace carve-up (floats)
    float* ws   = (float*)d_ws;
    float* xl1  = ws;                               // N*256
    float* xr1  = xl1 + (size_t)N_NODES * C1;       // N*256
    float* h1   = xr1 + (size_t)N_NODES * C1;       // N*256
    float* xl2  = h1  + (size_t)N_NODES * C1;       // N*64
    float* xr2  = xl2 + (size_t)N_NODES * HID;      // N*64
    float* h2   = xr2 + (size_t)N_NODES * HID;      // N*64
    float* lg1  = h2  + (size_t)N_NODES * HID;      // E*4
    float* lg2  = lg1 + (size_t)N_EDGES * HEADS;    // E
    float* amax1 = lg2 + (size_t)N_EDGES;           // N*4
    float* den1  = amax1 + (size_t)N_NODES * HEADS; // N*4
    float* amax2 = den1 + (size_t)N_NODES * HEADS;  // N
    float* den2  = amax2 + (size_t)N_NODES;         // N

    const float NEG_INF = -INFINITY;
    auto blocks = [](size_t n, int b) { return (unsigned)((n + b - 1) / b); };

    // init accumulators
    fill_f32<<<blocks((size_t)N_NODES * C1, 256), 256, 0, stream>>>(h1, 0.f, (size_t)N_NODES * C1);
    fill_f32<<<blocks((size_t)N_NODES * HID, 256), 256, 0, stream>>>(h2, 0.f, (size_t)N_NODES * HID);
    fill_f32<<<blocks((size_t)N_NODES * HEADS, 256), 256, 0, stream>>>(den1, 0.f, (size_t)N_NODES * HEADS);
    fill_f32<<<blocks((size_t)N_NODES, 256), 256, 0, stream>>>(den2, 0.f, (size_t)N_NODES);
    fill_f32<<<blocks((size_t)N_NODES * HEADS, 256), 256, 0, stream>>>(amax1, NEG_INF, (size_t)N_NODES * HEADS);
    fill_f32<<<blocks((size_t)N_NODES, 256), 256, 0, stream>>>(amax2, NEG_INF, (size_t)N_NODES);

    // ---- layer 1 ----
    dim3 g1(N_NODES / 16, C1 / 64);                 // 3125 x 4, 4 waves/block
    gemm_bias_wmma<<<g1, 128, 0, stream>>>(x, W1l, b1l, xl1, N_NODES, IN_F, C1);
    gemm_bias_wmma<<<g1, 128, 0, stream>>>(x, W1r, b1r, xr1, N_NODES, IN_F, C1);

    edge_logits1<<<N_EDGES / 8, 256, 0, stream>>>(xl1, xr1, eattr, src, dst,
                                                  W1e, b1e, att1, lg1, amax1);
    softmax_ex<<<blocks((size_t)N_EDGES * HEADS, 256), 256, 0, stream>>>(lg1, dst, amax1, den1, N_EDGES, HEADS);
    aggregate<<<blocks((size_t)N_EDGES * C1, 256), 256, 0, stream>>>(xl1, lg1, den1, src, dst, h1, N_EDGES, C1, HID);
    add_bias_relu<<<blocks((size_t)N_NODES * C1, 256), 256, 0, stream>>>(h1, bias1, C1, (size_t)N_NODES * C1);

    // ---- layer 2 ----
    dim3 g2(N_NODES / 16, HID / 64);                // 3125 x 1
    gemm_bias_wmma<<<g2, 128, 0, stream>>>(h1, W2l, b2l, xl2, N_NODES, C1, HID);
    gemm_bias_wmma<<<g2, 128, 0, stream>>>(h1, W2r, b2r, xr2, N_NODES, C1, HID);

    edge_logits2<<<N_EDGES / 8, 256, 0, stream>>>(xl2, xr2, eattr, src, dst,
                                                  W2e, b2e, att2, lg2, amax2);
    softmax_ex<<<blocks((size_t)N_EDGES, 256), 256, 0, stream>>>(lg2, dst, amax2, den2, N_EDGES, 1);
    aggregate<<<blocks((size_t)N_EDGES * HID, 256), 256, 0, stream>>>(xl2, lg2, den2, src, dst, h2, N_EDGES, HID, HID);
    add_bias_relu<<<blocks((size_t)N_NODES * HID, 256), 256, 0, stream>>>(h2, bias2, HID, (size_t)N_NODES * HID);

    // ---- edge MLP ----
    edge_mlp<<<N_EDGES / 16, 128, 0, stream>>>(h2, eattr, src, dst, Wm1, bm1, Wm2, bm2, out);
}